// EdgeDecoder_28681791603022
// MI455X (gfx1250) — compile-verified
//
#include <hip/hip_runtime.h>

#define N_NODES 100000
#define D 128
#define N_EDGES 600000

typedef __attribute__((ext_vector_type(2))) float v2f;
typedef __attribute__((ext_vector_type(8))) float v8f;

// Phase 1: per-node dot products via V_WMMA_F32_16X16X4_F32.
// One wave (32 lanes) per tile of 16 nodes; 100000/16 = 6250 tiles exactly.
// Computes s_row[n] = z[n,:] . W[0:128], s_col[n] = z[n,:] . W[128:256].
//
// Fragment layouts (ISA 7.12.2, wave32):
//   A (16x4 f32):  lane L<16: M=L, VGPR j holds K=k+j;  lanes 16-31: K=k+2+j
//   B (4x16 f32):  VGPR j: lanes 0-15 K=k+j, lanes 16-31 K=k+2+j; N = lane&15
//   C/D (16x16):   VGPR v: lanes 0-15 M=v, lanes 16-31 M=v+8; N = lane&15
__global__ __launch_bounds__(32)
void node_dots_wmma(const float* __restrict__ z, const float* __restrict__ W,
                    float* __restrict__ s_row, float* __restrict__ s_col) {
    const int tile = blockIdx.x;
    const int lane = threadIdx.x;   // 0..31
    const int m    = lane & 15;     // M (for A) / N (for B and C)
    const int hi   = lane >> 4;     // K sub-offset selector
    const int node = tile * 16 + m;
    const float* zrow = z + (size_t)node * D;

    // Branch-free B masking: only columns N=0 (w_row) and N=1 (w_col) are live.
    // Address is always in-bounds (W has 256 floats); sel zeroes dead columns.
    const float sel = (m < 2) ? 1.0f : 0.0f;
    const float* wp = W + (m & 1) * D;

    v8f acc = {};
    #pragma unroll 8
    for (int k = 0; k < D; k += 4) {
        const int ka = k + hi * 2;  // K index of VGPR0 for both A and B

        v2f a;                      // A[M=m][K=ka], A[M=m][K=ka+1]
        a.x = zrow[ka];
        a.y = zrow[ka + 1];

        v2f b;                      // B[K=ka][N=m], B[K=ka+1][N=m]
        b.x = wp[ka]     * sel;
        b.y = wp[ka + 1] * sel;

        // (neg_a, A, neg_b, B, c_mod, C, reuse_a, reuse_b)
        acc = __builtin_amdgcn_wmma_f32_16x16x4_f32(
            false, a, false, b, (short)0, acc, false, false);
    }

    // Extract: lane N=0 holds s_row for M = v + 8*hi, lane N=1 holds s_col.
    if (m < 2) {
        float* dst = (m == 0) ? s_row : s_col;
        const int base = tile * 16 + hi * 8;
        #pragma unroll
        for (int v = 0; v < 8; ++v)
            dst[base + v] = acc[v];
    }
}

// Phase 2: scores[e] = s_row[row[e]] + s_col[col[e]] + b
__global__ __launch_bounds__(256)
void edge_scores(const long long* __restrict__ ei,
                 const float* __restrict__ s_row,
                 const float* __restrict__ s_col,
                 const float* __restrict__ bias,
                 float* __restrict__ out) {
    const int e = blockIdx.x * blockDim.x + threadIdx.x;
    if (e >= N_EDGES) return;
    const long long r = ei[e];            // edge_index[0, e]
    const long long c = ei[N_EDGES + e];  // edge_index[1, e]
    out[e] = s_row[r] + s_col[c] + bias[0];
}

extern "C" void kernel_launch(void* const* d_in, const int* in_sizes, int n_in,
                              void* d_out, int out_size, void* d_ws, size_t ws_size,
                              hipStream_t stream) {
    const float*     z  = (const float*)d_in[0];
    const long long* ei = (const long long*)d_in[1];  // int64 edge_index (2 x N_EDGES)
    const float*     W  = (const float*)d_in[2];      // (1, 256)
    const float*     b  = (const float*)d_in[3];      // (1,)

    float* s_row = (float*)d_ws;          // N_NODES floats
    float* s_col = s_row + N_NODES;       // N_NODES floats (800 KB total in d_ws)

    node_dots_wmma<<<N_NODES / 16, 32, 0, stream>>>(z, W, s_row, s_col);
    edge_scores<<<(N_EDGES + 255) / 256, 256, 0, stream>>>(ei, s_row, s_col, b,
                                                           (float*)d_out);
}